// ScaleHash_35673998360637
// MI455X (gfx1250) — compile-verified
//
#include <hip/hip_runtime.h>
#include <cmath>

// ---------------------------------------------------------------------------
// Instant-NGP hashgrid forward, tuned for MI455X (gfx1250, wave32).
//
// Memory-bound gather kernel: 64 MB hash table is L2-resident (192 MB L2),
// ~537 MB of float2 gathers + 64 MB NT output stores. Optimization targets:
//   * branch-free, barrier-free per-thread code (instruction issue rate is
//     comparable to the L2 transfer time for this op)
//   * 8 independent global_load_b64 gathers in flight per thread
//   * v_and_or_b32-fusable index math ((h & MASK) | level<<19)
//   * NT stores so the write-once output never evicts the table from L2
// ---------------------------------------------------------------------------

#define NLEVELS   16
#define LOG2T     19
#define TSIZE     (1u << LOG2T)
#define TMASK     (TSIZE - 1u)

typedef float v2f __attribute__((ext_vector_type(2)));

struct Scales { float s[NLEVELS]; };

__global__ __launch_bounds__(256) void hashgrid_fwd_kernel(
    const float* __restrict__ in_tensor,   // [B,3] fp32
    const float* __restrict__ hash_table,  // [TSIZE*NLEVELS, 2] fp32
    float*       __restrict__ out,         // [B, NLEVELS*2] fp32
    Scales sc, int npoints)
{
    const int tid   = blockIdx.x * 256 + threadIdx.x;
    const int level = tid & (NLEVELS - 1);
    int point       = tid >> 4;
    // Branch-free clamp (no-op when npoints % 16 == 0, as with B = 524288):
    // duplicated work on the tail instead of EXEC-mask juggling.
    point = (point < npoints) ? point : (npoints - 1);

    // Direct coord loads: lanes 0-15 / 16-31 of a wave share a point each, so
    // these merge into ~2 cachelines per wave and hit L0 across levels.
    const float* __restrict__ pc = in_tensor + (size_t)point * 3u;
    const float x = pc[0];
    const float y = pc[1];
    const float z = pc[2];

    const float s  = sc.s[level];
    const float sx = x * s, sy = y * s, sz = z * s;

    // floor / ceil corners and fractional offsets (offset = scaled - floor)
    const float fxf = floorf(sx), fyf = floorf(sy), fzf = floorf(sz);
    const float ox = sx - fxf, oy = sy - fyf, oz = sz - fzf;

    const uint32_t fx = (uint32_t)(int)fxf;
    const uint32_t fy = (uint32_t)(int)fyf;
    const uint32_t fz = (uint32_t)(int)fzf;
    const uint32_t cx = (uint32_t)(int)ceilf(sx);
    const uint32_t cy = (uint32_t)(int)ceilf(sy);
    const uint32_t cz = (uint32_t)(int)ceilf(sz);

    // Spatial hash: h = x*1 ^ y*2654435761 ^ z*805459861.
    // Pre-multiply y/z partials, pre-XOR the 4 (y,z) combinations, then each
    // corner index is one v_xor + one v_and_or (mask | level-base, bit-disjoint).
    const uint32_t hyc = cy * 2654435761u;
    const uint32_t hyf = fy * 2654435761u;
    const uint32_t hzc = cz * 805459861u;
    const uint32_t hzf = fz * 805459861u;

    const uint32_t ycc = hyc ^ hzc;   // (cy, cz)
    const uint32_t yfc = hyf ^ hzc;   // (fy, cz)
    const uint32_t yff = hyf ^ hzf;   // (fy, fz)
    const uint32_t ycf = hyc ^ hzf;   // (cy, fz)

    const uint32_t base = (uint32_t)level << LOG2T;
    const v2f* __restrict__ tb = (const v2f*)hash_table;

#define IDX(HX, HYZ) ((((HX) ^ (HYZ)) & TMASK) | base)
    // All 8 gathers issued back-to-back -> 8 loads in flight, one wait.
    const v2f f0 = tb[IDX(cx, ycc)];
    const v2f f1 = tb[IDX(cx, yfc)];
    const v2f f2 = tb[IDX(fx, yfc)];
    const v2f f3 = tb[IDX(fx, ycc)];
    const v2f f4 = tb[IDX(cx, ycf)];
    const v2f f5 = tb[IDX(cx, yff)];
    const v2f f6 = tb[IDX(fx, yff)];
    const v2f f7 = tb[IDX(fx, ycf)];
#undef IDX

    // Trilinear interpolation, identical operand order to the reference.
    const float mx = 1.0f - ox, my = 1.0f - oy, mz = 1.0f - oz;
    const v2f f03   = f0 * ox + f3 * mx;
    const v2f f12   = f1 * ox + f2 * mx;
    const v2f f56   = f5 * ox + f6 * mx;
    const v2f f47   = f4 * ox + f7 * mx;
    const v2f f0312 = f03 * oy + f12 * my;
    const v2f f4756 = f47 * oy + f56 * my;
    const v2f enc   = f0312 * oz + f4756 * mz;

    // 16 consecutive lanes (one point) write 16 consecutive float2 ->
    // perfectly coalesced 256B/wave NT stores (write-once data; keep L2 for
    // the hash table).
    v2f* __restrict__ o = (v2f*)out + (size_t)point * NLEVELS + level;
    __builtin_nontemporal_store(enc, o);
}

extern "C" void kernel_launch(void* const* d_in, const int* in_sizes, int n_in,
                              void* d_out, int out_size, void* d_ws, size_t ws_size,
                              hipStream_t stream) {
    const float* in_tensor  = (const float*)d_in[0];
    const float* hash_table = (const float*)d_in[1];
    float* out              = (float*)d_out;

    const int npoints = in_sizes[0] / 3;

    // Replicate numpy's float64 scaling computation exactly on the host:
    // growth = exp((log(4096)-log(16))/15); scale_l = floor(16 * growth^l)
    Scales sc;
    const double growth = std::exp((std::log(4096.0) - std::log(16.0)) /
                                   (double)(NLEVELS - 1));
    for (int l = 0; l < NLEVELS; ++l) {
        sc.s[l] = (float)std::floor(16.0 * std::pow(growth, (double)l));
    }

    const long long threads = (long long)npoints * NLEVELS;
    const int blocks = (int)((threads + 255) / 256);
    hashgrid_fwd_kernel<<<blocks, 256, 0, stream>>>(
        in_tensor, hash_table, out, sc, npoints);
}